// MultiHeadAttention_53025666236969
// MI455X (gfx1250) — compile-verified
//
#include <hip/hip_runtime.h>
#include <hip/hip_bf16.h>

// Shapes from the reference
#define BH   16        // B*H
#define NSER 32        // N
#define DQK  3072      // T*D_HEAD
#define TFD  3072      // T*D_HEAD (output inner dim, contiguous)

typedef __attribute__((ext_vector_type(2))) float v2f;
typedef __attribute__((ext_vector_type(4))) float v4f;
typedef __attribute__((ext_vector_type(8))) float v8f;

// -------------------------------------------------------------------------
// Phase 1: per (b,h) compute S = Q K^T / sqrt(DQK), mask, softmax -> attn.
// 16 blocks x 128 threads (4 waves). Wave w computes 16x16 tile (mi,ni) of
// the 32x32 score matrix via V_WMMA_F32_16X16X4_F32, fp32-exact.
// 4 independent accumulators hide the WMMA RAW latency (768 -> 4x192 chain).
// -------------------------------------------------------------------------
__global__ __launch_bounds__(128) void mha_scores_softmax(
    const float* __restrict__ Q, const float* __restrict__ K,
    const int* __restrict__ mask, float* __restrict__ attn)
{
    __shared__ float S[NSER][NSER + 1];   // +1 pad: conflict-free row reads

    const int bh   = blockIdx.x;          // 0..15
    const int tid  = threadIdx.x;
    const int wave = tid >> 5;
    const int lane = tid & 31;
    const int half = lane >> 4;           // 0: K={0,1}, 1: K={2,3}
    const int m16  = lane & 15;           // M (or N) index within tile
    const int mi   = wave >> 1;           // query-row tile
    const int ni   = wave & 1;            // key-col tile

    // A fragment source: Q row (mi*16+m16), cols k0 + 2*half + {0,1}
    // B fragment source: K row (ni*16+m16) (B = K^T), same col pattern
    const float* qrow = Q + ((size_t)bh * NSER + mi * 16 + m16) * DQK + 2 * half;
    const float* krow = K + ((size_t)bh * NSER + ni * 16 + m16) * DQK + 2 * half;

    v8f acc0 = {}, acc1 = {}, acc2 = {}, acc3 = {};
    for (int k0 = 0; k0 < DQK; k0 += 16) {
        v2f a0 = *(const v2f*)(qrow + k0);
        v2f b0 = *(const v2f*)(krow + k0);
        v2f a1 = *(const v2f*)(qrow + k0 + 4);
        v2f b1 = *(const v2f*)(krow + k0 + 4);
        v2f a2 = *(const v2f*)(qrow + k0 + 8);
        v2f b2 = *(const v2f*)(krow + k0 + 8);
        v2f a3 = *(const v2f*)(qrow + k0 + 12);
        v2f b3 = *(const v2f*)(krow + k0 + 12);
        acc0 = __builtin_amdgcn_wmma_f32_16x16x4_f32(false, a0, false, b0,
                                                     (short)0, acc0, false, false);
        acc1 = __builtin_amdgcn_wmma_f32_16x16x4_f32(false, a1, false, b1,
                                                     (short)0, acc1, false, false);
        acc2 = __builtin_amdgcn_wmma_f32_16x16x4_f32(false, a2, false, b2,
                                                     (short)0, acc2, false, false);
        acc3 = __builtin_amdgcn_wmma_f32_16x16x4_f32(false, a3, false, b3,
                                                     (short)0, acc3, false, false);
    }
    v8f acc = acc0 + acc1 + acc2 + acc3;

    // 1 / (sqrt(T*D_HEAD) * TAU), TAU = 1
    const float inv_scale = 1.0f / 55.42562584220407f;

    // C/D layout: VGPR r -> M = r (lanes 0-15) or M = 8+r (lanes 16-31),
    // N = lane & 15.
    #pragma unroll
    for (int r = 0; r < 8; ++r) {
        const int row = mi * 16 + half * 8 + r;   // query index i
        const int col = ni * 16 + m16;            // key index j
        float v = acc[r] * inv_scale;
        if (mask[((size_t)bh * NSER + row) * NSER + col] == 0)
            v = -__builtin_inff();
        S[row][col] = v;
    }
    __syncthreads();

    // Softmax over j: one thread per row.
    if (tid < NSER) {
        float m = -__builtin_inff();
        #pragma unroll
        for (int c = 0; c < NSER; ++c) m = fmaxf(m, S[tid][c]);
        float s = 0.0f;
        #pragma unroll
        for (int c = 0; c < NSER; ++c) {
            float ev = expf(S[tid][c] - m);
            S[tid][c] = ev;
            s += ev;
        }
        const float invs = 1.0f / s;
        float* arow = attn + ((size_t)bh * NSER + tid) * NSER;
        #pragma unroll
        for (int c = 0; c < NSER; ++c) arow[c] = S[tid][c] * invs;
    }
}

// -------------------------------------------------------------------------
// Phase 2: out[bh,i,tf] = sum_j attn[bh,i,j] * V[bh,j,i,tf].
// Streams 201 MB of V exactly once -> HBM-bound. One block per (b,h,i):
// 512 blocks x 256 threads; each thread owns 3 float4 slices of the 3072
// contiguous tf elements. Non-temporal loads keep V out of L2 (V > 192MB).
// -------------------------------------------------------------------------
__global__ __launch_bounds__(256) void mha_apply_v(
    const float* __restrict__ V, const float* __restrict__ attn,
    float* __restrict__ out)
{
    const int bhi = blockIdx.x;           // 0..511
    const int i   = bhi & (NSER - 1);
    const int bh  = bhi >> 5;
    const int tid = threadIdx.x;

    const float* arow = attn + ((size_t)bh * NSER + i) * NSER;
    // V flat index: ((bh*N + j)*N + i) * TFD ; j-stride = N*TFD floats
    const float*  vbase   = V + ((size_t)bh * NSER * NSER + i) * TFD;
    const size_t  jstride = (size_t)NSER * TFD;

    v4f acc0 = {}, acc1 = {}, acc2 = {};

    #pragma unroll 4
    for (int j = 0; j < NSER; ++j) {
        const float w = arow[j];
        const v4f* vp = (const v4f*)(vbase + (size_t)j * jstride);
        v4f x0 = __builtin_nontemporal_load(vp + tid);
        v4f x1 = __builtin_nontemporal_load(vp + 256 + tid);
        v4f x2 = __builtin_nontemporal_load(vp + 512 + tid);
        acc0 += w * x0;
        acc1 += w * x1;
        acc2 += w * x2;
    }

    v4f* ob = (v4f*)(out + ((size_t)bh * NSER + i) * TFD);
    ob[tid]       = acc0;
    ob[256 + tid] = acc1;
    ob[512 + tid] = acc2;
}

// -------------------------------------------------------------------------
extern "C" void kernel_launch(void* const* d_in, const int* in_sizes, int n_in,
                              void* d_out, int out_size, void* d_ws, size_t ws_size,
                              hipStream_t stream) {
    const float* Q    = (const float*)d_in[0];
    const float* K    = (const float*)d_in[1];
    const float* V    = (const float*)d_in[2];
    const int*   mask = (const int*)d_in[3];
    float*       out  = (float*)d_out;
    float*       attn = (float*)d_ws;     // B*H*N*N floats = 64 KB

    mha_scores_softmax<<<BH, 128, 0, stream>>>(Q, K, mask, attn);
    mha_apply_v<<<BH * NSER, 256, 0, stream>>>(V, attn, out);
}